// ModelNew_11888469475803
// MI455X (gfx1250) — compile-verified
//
#include <hip/hip_runtime.h>

typedef __attribute__((ext_vector_type(2))) float v2f;
typedef __attribute__((ext_vector_type(8))) float v8f;

#define CDIM   1920
#define NBATCH 256
#define HW     196
#define KOUT   1000
#define BN_EPS 1e-5f

// ---------------------------------------------------------------------------
// Stage 1: fused BN-affine + ReLU + global average pool.
// One wave32 per (n, c) pair; 784 contiguous bytes per task -> float4 loads.
// HBM-bound: ~385 MB of x traffic, floor ~16.5 us at 23.3 TB/s.
// ---------------------------------------------------------------------------
__global__ __launch_bounds__(256) void bn_relu_pool_kernel(
    const float* __restrict__ x,
    const float* __restrict__ bn_w,
    const float* __restrict__ bn_b,
    const float* __restrict__ rmean,
    const float* __restrict__ rvar,
    float* __restrict__ pooled)
{
    const int wid  = blockIdx.x * (256 / 32) + (threadIdx.x >> 5);
    const int lane = threadIdx.x & 31;
    if (wid >= NBATCH * CDIM) return;

    const int c = wid % CDIM;
    const float s = bn_w[c] * rsqrtf(rvar[c] + BN_EPS);   // fold running stats
    const float t = bn_b[c] - s * rmean[c];

    const float4* v = (const float4*)(x + (size_t)wid * HW);  // 49 float4, 16B aligned
    float sum = 0.0f;
    for (int i = lane; i < HW / 4; i += 32) {
        float4 q = v[i];
        sum += fmaxf(fmaf(q.x, s, t), 0.0f);
        sum += fmaxf(fmaf(q.y, s, t), 0.0f);
        sum += fmaxf(fmaf(q.z, s, t), 0.0f);
        sum += fmaxf(fmaf(q.w, s, t), 0.0f);
    }

    #pragma unroll
    for (int off = 16; off > 0; off >>= 1)
        sum += __shfl_down(sum, off, 32);

    if (lane == 0) pooled[wid] = sum * (1.0f / (float)HW);
}

// ---------------------------------------------------------------------------
// Stage 2: logits[n,k] = pooled[n,:] . weight[k,:] + bias[k]
// fp32 WMMA: V_WMMA_F32_16X16X4_F32, one wave per 16x16 output tile.
// A (16x4 fp32): lanes 0-15 -> K={0,1}, lanes 16-31 -> K={2,3} (2 VGPRs).
// B (4x16 fp32): N striped across lanes, same half-wave K split.
// C/D (16x16 f32): VGPR j -> row M=j (lanes 0-15) / M=j+8 (lanes 16-31).
// ---------------------------------------------------------------------------
__global__ __launch_bounds__(256) void gemm_wmma_kernel(
    const float* __restrict__ pooled,   // [NBATCH, CDIM]
    const float* __restrict__ weight,   // [KOUT, CDIM]
    const float* __restrict__ bias,     // [KOUT]
    float* __restrict__ out)            // [NBATCH, KOUT]
{
    const int NT = (KOUT + 15) / 16;                    // 63 N-tiles (last padded)
    const int wid  = blockIdx.x * (256 / 32) + (threadIdx.x >> 5);
    const int lane = threadIdx.x & 31;

    const int mt = wid / NT;            // 0..15
    const int nt = wid - mt * NT;       // 0..62
    const int lr = lane & 15;
    const int hi = lane >> 4;           // half-wave: selects K pair {0,1} vs {2,3}

    const int arow = mt * 16 + lr;                      // A row this lane feeds
    const int col  = nt * 16 + lr;                      // output column
    const int colc = (col < KOUT) ? col : (KOUT - 1);   // clamp for safe loads

    const float* ap = pooled + (size_t)arow * CDIM + hi * 2;
    const float* bp = weight + (size_t)colc * CDIM + hi * 2;

    v8f acc = {};
    #pragma unroll 8
    for (int kk = 0; kk < CDIM; kk += 4) {
        v2f a = *(const v2f*)(ap + kk);
        v2f b = *(const v2f*)(bp + kk);
        // D = A(16x4) x B(4x16) + C, exact fp32
        acc = __builtin_amdgcn_wmma_f32_16x16x4_f32(
            /*neg_a=*/false, a, /*neg_b=*/false, b,
            /*c_mod=*/(short)0, acc, /*reuse_a=*/false, /*reuse_b=*/false);
    }

    const float bv = bias[colc];
    if (col < KOUT) {                   // divergence only after all WMMAs issued
        const int mbase = mt * 16 + hi * 8;
        #pragma unroll
        for (int j = 0; j < 8; ++j)
            out[(size_t)(mbase + j) * KOUT + col] = acc[j] + bv;
    }
}

// ---------------------------------------------------------------------------
extern "C" void kernel_launch(void* const* d_in, const int* in_sizes, int n_in,
                              void* d_out, int out_size, void* d_ws, size_t ws_size,
                              hipStream_t stream) {
    const float* x      = (const float*)d_in[0];
    const float* bn_w   = (const float*)d_in[1];
    const float* bn_b   = (const float*)d_in[2];
    const float* rmean  = (const float*)d_in[3];
    const float* rvar   = (const float*)d_in[4];
    const float* weight = (const float*)d_in[5];
    const float* bias   = (const float*)d_in[6];
    float* out    = (float*)d_out;
    float* pooled = (float*)d_ws;       // NBATCH*CDIM floats = ~1.97 MB scratch

    // 256*1920 = 491520 (n,c) tasks, 8 waves/block -> 61440 blocks
    bn_relu_pool_kernel<<<(NBATCH * CDIM) / 8, 256, 0, stream>>>(
        x, bn_w, bn_b, rmean, rvar, pooled);

    // 16 M-tiles * 63 N-tiles = 1008 waves, 8 waves/block -> 126 blocks
    gemm_wmma_kernel<<<126, 256, 0, stream>>>(pooled, weight, bias, out);
}